// _x_to_y_16303695856162
// MI455X (gfx1250) — compile-verified
//
#include <hip/hip_runtime.h>

// CDNA5 / gfx1250, wave32. FP32 WMMA (16x16x4) MLP with transposed activations.
// - invariants per lane (VALU), MLP layers on the WMMA units (A = W^T, fp32)
// - half-wave swaps via v_permlanex16_b32 (VALU, no LDS)
// - softplus via hardware v_exp_f32 / v_log_f32
// - wave-granular grid-stride loop: weights/biases preloaded to registers once
//   per wave, each wave processes many 32-sample chunks (EXEC stays all-ones).

typedef __attribute__((ext_vector_type(2))) float v2f;
typedef __attribute__((ext_vector_type(8))) float v8f;

__device__ __forceinline__ float swap16(float x) {
  // exchange lower/upper half-wave lanes (lane L <-> L^16) via v_permlanex16:
  // each lane i of a 16-lane row fetches lane i of the other row (nibble i = i).
  int xi = __builtin_bit_cast(int, x);
  int r = __builtin_amdgcn_permlanex16(xi, xi, 0x76543210, 0xfedcba98,
                                       /*fi=*/false, /*bound_ctrl=*/false);
  return __builtin_bit_cast(float, r);
}

__device__ __forceinline__ float sel(bool lo, float a, float b) { return lo ? a : b; }

__device__ __forceinline__ v8f wmma4(v2f a, v2f b, v8f c) {
  // D(16x16,f32) = A(16x4,f32) * B(4x16,f32) + C
  return __builtin_amdgcn_wmma_f32_16x16x4_f32(false, a, false, b, (short)0, c,
                                               false, false);
}

__device__ __forceinline__ float softplus_f(float x) {
  // softplus(x) = max(x,0) + log(1 + exp(-|x|)), hardware exp2/log2 based.
  const float LOG2E = 1.4426950408889634f;
  const float LN2   = 0.6931471805599453f;
  float t = __builtin_amdgcn_exp2f(-LOG2E * fabsf(x));   // exp(-|x|) in (0,1]
  float l = __builtin_amdgcn_logf(1.0f + t);             // log2(1+t), arg in (1,2]
  return fmaxf(x, 0.0f) + LN2 * l;
}

__device__ __forceinline__ v8f act(v8f x, const float* bv) {
  v8f r;
#pragma unroll
  for (int j = 0; j < 8; ++j) r[j] = softplus_f(x[j] + bv[j]);
  return r;
}

// One 16->16 layer on one 16-sample tile.
// x is T = act[feature=row, sample=col] in C/D layout (VGPR j: row j lanes 0-15,
// row j+8 lanes 16-31). Build the four K=4 B-operand chunks with half-wave
// swaps + selects, accumulate 4 WMMAs against preloaded A = W^T chunks.
__device__ __forceinline__ v8f layer16(v8f x, const v2f* w, bool lo) {
  float sx0 = swap16(x[0]), sx1 = swap16(x[1]), sx2 = swap16(x[2]), sx3 = swap16(x[3]);
  float sx4 = swap16(x[4]), sx5 = swap16(x[5]), sx6 = swap16(x[6]), sx7 = swap16(x[7]);
  v8f acc = {};
  v2f b;
  // chunk 0: rows 0,2 | 1,3
  b[0] = sel(lo, x[0], sx2); b[1] = sel(lo, x[1], sx3);
  acc = wmma4(w[0], b, acc);
  // chunk 1: rows 4,6 | 5,7
  b[0] = sel(lo, x[4], sx6); b[1] = sel(lo, x[5], sx7);
  acc = wmma4(w[1], b, acc);
  // chunk 2: rows 8,10 | 9,11
  b[0] = sel(lo, sx0, x[2]); b[1] = sel(lo, sx1, x[3]);
  acc = wmma4(w[2], b, acc);
  // chunk 3: rows 12,14 | 13,15
  b[0] = sel(lo, sx4, x[6]); b[1] = sel(lo, sx5, x[7]);
  acc = wmma4(w[3], b, acc);
  return acc;
}

__global__ __launch_bounds__(256)
void hyperelastic_mlp_kernel(const float* __restrict__ Fs,
                             const float* __restrict__ W1, const float* __restrict__ b1,
                             const float* __restrict__ W2, const float* __restrict__ b2,
                             const float* __restrict__ W3, const float* __restrict__ b3,
                             const float* __restrict__ W4, const float* __restrict__ b4,
                             float* __restrict__ out, unsigned n) {
  const unsigned lane = threadIdx.x & 31u;
  const bool lo = lane < 16u;
  const unsigned m = lane & 15u;       // M index for A-operand / out-feature col
  const unsigned kh = lane >> 4;       // 0: lanes 0-15, 1: lanes 16-31

  // ---- preload weights as WMMA A-operands: A = W^T (M=out-feat, K=in-feat).
  // A-chunk k, vgpr v, lane: W[(4k + v + 2*kh)][m], W row-major [in][out].
  v2f aW1[2];
#pragma unroll
  for (int k = 0; k < 2; ++k)
#pragma unroll
    for (int v = 0; v < 2; ++v) {
      unsigned r = 4u * k + (unsigned)v + 2u * kh;
      aW1[k][v] = (r < 6u) ? W1[r * 16u + m] : 0.0f;   // K padded 6->8
    }
  v2f aW2[4], aW3[4];
#pragma unroll
  for (int k = 0; k < 4; ++k)
#pragma unroll
    for (int v = 0; v < 2; ++v) {
      unsigned r = 4u * k + (unsigned)v + 2u * kh;
      aW2[k][v] = W2[r * 16u + m];
      aW3[k][v] = W3[r * 16u + m];
    }
  // biases / final weights per accumulator VGPR j (row j for lo, j+8 for hi)
  float bv1[8], bv2[8], bv3[8], w4v[8];
#pragma unroll
  for (int j = 0; j < 8; ++j) {
    bv1[j] = b1[j + 8u * kh];
    bv2[j] = b2[j + 8u * kh];
    bv3[j] = b3[j + 8u * kh];
    w4v[j] = W4[j + 8u * kh];
  }
  const float bias4 = b4[0];

  // ---- wave-granular grid-stride loop: 32 samples per iteration.
  // Loop bound is wave-uniform -> EXEC is all-ones at every WMMA.
  const unsigned wstride = gridDim.x * blockDim.x;           // samples per step
  for (unsigned wbase = blockIdx.x * blockDim.x + (threadIdx.x & ~31u);
       wbase < n; wbase += wstride) {
    // ---- per-lane invariants for sample s = wbase + lane
    const unsigned s = wbase + lane;
    const float* F = Fs + (size_t)s * 9u;
    float f[9];
#pragma unroll
    for (int i = 0; i < 9; ++i) f[i] = F[i];

    // C = F^T F (right Cauchy-Green, symmetric)
    float C00 = f[0]*f[0] + f[3]*f[3] + f[6]*f[6];
    float C11 = f[1]*f[1] + f[4]*f[4] + f[7]*f[7];
    float C22 = f[2]*f[2] + f[5]*f[5] + f[8]*f[8];
    float C01 = f[0]*f[1] + f[3]*f[4] + f[6]*f[7];
    float C02 = f[0]*f[2] + f[3]*f[5] + f[6]*f[8];
    float C12 = f[1]*f[2] + f[4]*f[5] + f[7]*f[8];
    float I1 = C00 + C11 + C22;
    float J  = f[0]*(f[4]*f[8] - f[5]*f[7])
             - f[1]*(f[3]*f[8] - f[5]*f[6])
             + f[2]*(f[3]*f[7] - f[4]*f[6]);
    // diagonal of cof(C) = det(C)*inv(C) for symmetric C
    float K0 = C11*C22 - C12*C12;
    float K1 = C00*C22 - C02*C02;
    float K2 = C00*C11 - C01*C01;
    float I2  = K0 + K1 + K2;
    float I7  = C00*C00 + C11*C11 + C22*C22;
    float I11 = K0*K0 + K1*K1 + K2*K2;

    const float inv0 = I1, inv1 = I2, inv2 = J, inv3 = -J, inv4 = I7, inv5 = I11;

    // ---- layer 1: two 16-sample tiles (A: samples wbase..+15, B: +16..+31)
    float si0 = swap16(inv0), si1 = swap16(inv1), si2 = swap16(inv2);
    float si3 = swap16(inv3), si4 = swap16(inv4), si5 = swap16(inv5);

    v8f xA = {}, xB = {};
    v2f bb;
    // tile A, chunk 0: rows 0,2 | 1,3
    bb[0] = sel(lo, inv0, si2); bb[1] = sel(lo, inv1, si3);
    xA = wmma4(aW1[0], bb, xA);
    // tile A, chunk 1: rows 4,(6=pad) | 5,(7=pad)
    bb[0] = sel(lo, inv4, 0.0f); bb[1] = sel(lo, inv5, 0.0f);
    xA = wmma4(aW1[1], bb, xA);
    // tile B, chunk 0
    bb[0] = sel(lo, si0, inv2); bb[1] = sel(lo, si1, inv3);
    xB = wmma4(aW1[0], bb, xB);
    // tile B, chunk 1
    bb[0] = sel(lo, si4, 0.0f); bb[1] = sel(lo, si5, 0.0f);
    xB = wmma4(aW1[1], bb, xB);

    xA = act(xA, bv1);
    xB = act(xB, bv1);

    // ---- layers 2 and 3 (16->16, WMMA)
    xA = act(layer16(xA, aW2, lo), bv2);
    xB = act(layer16(xB, aW2, lo), bv2);
    xA = act(layer16(xA, aW3, lo), bv3);
    xB = act(layer16(xB, aW3, lo), bv3);

    // ---- layer 4: 16->1 dot + cross-half reduce
    float sA = 0.0f, sB = 0.0f;
#pragma unroll
    for (int j = 0; j < 8; ++j) {
      sA += xA[j] * w4v[j];
      sB += xB[j] * w4v[j];
    }
    sA += swap16(sA);   // all lanes: full sum for sample (lane&15) of tile A
    sB += swap16(sB);   // ... of tile B

    // lane L<16 -> tile A sample wbase+L; lane L>=16 -> tile B sample wbase+L
    out[s] = sel(lo, sA, sB) + bias4;
  }
}

extern "C" void kernel_launch(void* const* d_in, const int* in_sizes, int n_in,
                              void* d_out, int out_size, void* d_ws, size_t ws_size,
                              hipStream_t stream) {
  const float* Fs = (const float*)d_in[0];
  const float* W1 = (const float*)d_in[1];
  const float* b1 = (const float*)d_in[2];
  const float* W2 = (const float*)d_in[3];
  const float* b2 = (const float*)d_in[4];
  const float* W3 = (const float*)d_in[5];
  const float* b3 = (const float*)d_in[6];
  const float* W4 = (const float*)d_in[7];
  const float* b4 = (const float*)d_in[8];
  float* out = (float*)d_out;

  unsigned n = (unsigned)(in_sizes[0] / 9);   // number of samples (2,097,152)
  dim3 block(256);
  // ~8 chunks of 32 samples per wave: amortizes the per-wave weight preload
  // while keeping ~8k waves in flight for latency hiding.
  unsigned blocks = (n + 2047u) / 2048u;
  if (blocks == 0) blocks = 1;
  hyperelastic_mlp_kernel<<<dim3(blocks), block, 0, stream>>>(
      Fs, W1, b1, W2, b2, W3, b3, W4, b4, out, n);
}